// PlasticLSTM_14508399526005
// MI455X (gfx1250) — compile-verified
//
#include <hip/hip_runtime.h>
#include <hip/hip_bf16.h>
#include <math.h>

typedef float v2f __attribute__((ext_vector_type(2)));
typedef float v4f __attribute__((ext_vector_type(4)));
typedef float v8f __attribute__((ext_vector_type(8)));

#define BATCH 256
#define ISZ   256
#define HSZ   512
#define BH    (BATCH * HSZ)
#define CLIPV 2.0f
#define NCHUNK 4            // h-dimension split for the hebb reduction

__device__ __forceinline__ float sigmoidf_(float x) {
    return 1.0f / (1.0f + __expf(-x));
}

// ---------------------------------------------------------------------------
// Kernel 1: all small GEMMs via V_WMMA_F32_16X16X4_F32.
// jobs 0..2: sigmoid(x @ x2g_w.T + x2g_b + h0 @ h2g_w.T + h2g_b)   (f, i, o)
// job  3   : x @ x2c_w.T + x2c_b                                    (cpre)
// job  4   : h0 @ w                                                 (hw)
// One wave per 16x16 output tile. 5 jobs * (256/16)*(512/16) = 2560 waves.
// All operands stay L2-resident (~6 MB working set) so no LDS tiling needed.
// ---------------------------------------------------------------------------
__global__ __launch_bounds__(256) void gate_gemm_kernel(
    const float* __restrict__ x,    const float* __restrict__ h0,
    const float* __restrict__ wmat,
    const float* __restrict__ h2f_w, const float* __restrict__ h2f_b,
    const float* __restrict__ h2i_w, const float* __restrict__ h2i_b,
    const float* __restrict__ h2o_w, const float* __restrict__ h2o_b,
    const float* __restrict__ x2f_w, const float* __restrict__ x2f_b,
    const float* __restrict__ x2i_w, const float* __restrict__ x2i_b,
    const float* __restrict__ x2o_w, const float* __restrict__ x2o_b,
    const float* __restrict__ x2c_w, const float* __restrict__ x2c_b,
    float* __restrict__ ws)
{
    const int TM = BATCH / 16;   // 16 tiles in M
    const int TN = HSZ / 16;     // 32 tiles in N
    int wave = blockIdx.x * (blockDim.x >> 5) + (threadIdx.x >> 5);
    int lane = threadIdx.x & 31;
    int job  = wave / (TM * TN);
    int t    = wave % (TM * TN);
    int m0   = (t / TN) * 16;
    int n0   = (t % TN) * 16;
    int half = lane >> 4;        // 0: K{0,1} / M 0..7 ; 1: K{2,3} / M 8..15
    int lr   = lane & 15;

    const float* xw = nullptr; const float* xb = nullptr;
    const float* hw = nullptr; const float* hb = nullptr;
    if      (job == 0) { xw = x2f_w; xb = x2f_b; hw = h2f_w; hb = h2f_b; }
    else if (job == 1) { xw = x2i_w; xb = x2i_b; hw = h2i_w; hb = h2i_b; }
    else if (job == 2) { xw = x2o_w; xb = x2o_b; hw = h2o_w; hb = h2o_b; }
    else if (job == 3) { xw = x2c_w; xb = x2c_b; }

    v8f acc = {};

    if (job <= 3) {
        // C += x(256x256) * x2?_w.T   -> B[k][n] = xw[n*ISZ + k]
        const float* arow = x  + (size_t)(m0 + lr) * ISZ;
        const float* brow = xw + (size_t)(n0 + lr) * ISZ;
        for (int k = 0; k < ISZ; k += 4) {
            int ka = k + (half << 1);
            v2f a = { arow[ka], arow[ka + 1] };
            v2f b = { brow[ka], brow[ka + 1] };
            acc = __builtin_amdgcn_wmma_f32_16x16x4_f32(
                false, a, false, b, (short)0, acc, false, false);
        }
    }
    if (job <= 2) {
        // C += h0(256x512) * h2?_w.T  -> B[k][n] = hw[n*HSZ + k]
        const float* arow = h0 + (size_t)(m0 + lr) * HSZ;
        const float* brow = hw + (size_t)(n0 + lr) * HSZ;
        for (int k = 0; k < HSZ; k += 4) {
            int ka = k + (half << 1);
            v2f a = { arow[ka], arow[ka + 1] };
            v2f b = { brow[ka], brow[ka + 1] };
            acc = __builtin_amdgcn_wmma_f32_16x16x4_f32(
                false, a, false, b, (short)0, acc, false, false);
        }
    }
    if (job == 4) {
        // C = h0 @ w, w row-major [h][k] -> B[k][n] = w[k*HSZ + n]
        const float* arow = h0 + (size_t)(m0 + lr) * HSZ;
        for (int k = 0; k < HSZ; k += 4) {
            int ka = k + (half << 1);
            v2f a = { arow[ka], arow[ka + 1] };
            v2f b = { wmat[(size_t)ka * HSZ + n0 + lr],
                      wmat[(size_t)(ka + 1) * HSZ + n0 + lr] };
            acc = __builtin_amdgcn_wmma_f32_16x16x4_f32(
                false, a, false, b, (short)0, acc, false, false);
        }
    }

    // C/D layout: VGPR r -> M = r + 8*half, N = lane%16
    float* outbase = ws + (size_t)job * BH;
    int   n    = n0 + lr;
    float bias = 0.0f;
    if (job <= 2)      bias = xb[n] + hb[n];
    else if (job == 3) bias = xb[n];
#pragma unroll
    for (int r = 0; r < 8; ++r) {
        int   m = m0 + r + (half << 3);
        float v = acc[r] + bias;
        if (job <= 2) v = sigmoidf_(v);
        outbase[(size_t)m * HSZ + n] = v;
    }
}

// ---------------------------------------------------------------------------
// Kernel 2a: partial r[b][k] = sum_{h in chunk} h0[b][h]*hebb[b][h][k]
// Streams the 268 MB hebb tensor once with non-temporal B128 loads.
// Grid (BATCH, NCHUNK) = 1024 blocks, 128 threads (thread = one float4 slab).
// ---------------------------------------------------------------------------
__global__ __launch_bounds__(128) void plastic_partial_kernel(
    const float* __restrict__ h0, const float* __restrict__ hebb,
    float* __restrict__ ws_part)
{
    __shared__ float sh[HSZ / NCHUNK];           // 128 h0 values for this chunk
    int b     = blockIdx.x;
    int chunk = blockIdx.y;
    int tid   = threadIdx.x;                     // 0..127 -> k = 4*tid..4*tid+3
    int h0off = chunk * (HSZ / NCHUNK);
    sh[tid] = h0[(size_t)b * HSZ + h0off + tid];
    __syncthreads();

    const float* hrow = hebb + ((size_t)b * HSZ + h0off) * HSZ;
    v4f acc = {};
    for (int i = 0; i < HSZ / NCHUNK; ++i) {
        float hv = sh[i];
        v4f v = __builtin_nontemporal_load(
            (const v4f*)(hrow + (size_t)i * HSZ) + tid);
        acc += hv * v;
    }
    *((v4f*)(ws_part + ((size_t)chunk * BATCH + b) * HSZ) + tid) = acc;
}

// ---------------------------------------------------------------------------
// Kernel 2b: combine partials, then
//   itc = tanh(cpre + hw + alpha[k]*r); cell = f*c0 + i*itc; hactiv = o*tanh(cell)
// ---------------------------------------------------------------------------
__global__ __launch_bounds__(128) void plastic_combine_kernel(
    const float* __restrict__ c0, const float* __restrict__ alpha,
    const float* __restrict__ ws, const float* __restrict__ ws_part,
    float* __restrict__ ws_itc, float* __restrict__ hactiv,
    float* __restrict__ cell)
{
    int b   = blockIdx.x;
    int tid = threadIdx.x;                       // k = 4*tid..4*tid+3
    size_t kbase = (size_t)b * HSZ + 4 * tid;

    v4f r = {};
#pragma unroll
    for (int c = 0; c < NCHUNK; ++c)
        r += *((const v4f*)(ws_part + ((size_t)c * BATCH + b) * HSZ) + tid);

    v4f al  = *((const v4f*)(alpha) + tid);
    v4f f4  = *(const v4f*)(ws + 0 * (size_t)BH + kbase);
    v4f i4  = *(const v4f*)(ws + 1 * (size_t)BH + kbase);
    v4f o4  = *(const v4f*)(ws + 2 * (size_t)BH + kbase);
    v4f cp4 = *(const v4f*)(ws + 3 * (size_t)BH + kbase);
    v4f hw4 = *(const v4f*)(ws + 4 * (size_t)BH + kbase);
    v4f c04 = *(const v4f*)(c0 + kbase);

    v4f itc4, cl4, ha4;
#pragma unroll
    for (int c = 0; c < 4; ++c) {
        float pre = cp4[c] + hw4[c] + al[c] * r[c];
        float itc = tanhf(pre);
        float cl  = f4[c] * c04[c] + i4[c] * itc;
        float ha  = o4[c] * tanhf(cl);
        itc4[c] = itc; cl4[c] = cl; ha4[c] = ha;
    }
    *(v4f*)(ws_itc + kbase) = itc4;
    *(v4f*)(cell   + kbase) = cl4;
    *(v4f*)(hactiv + kbase) = ha4;
}

// ---------------------------------------------------------------------------
// Kernel 3: m[b] = tanh(hactiv[b] . h2mod_w + h2mod_b)
//           g[b][k] = (m*modfanout_w[k] + modfanout_b[k]) * itc[b][k]
// ---------------------------------------------------------------------------
__global__ __launch_bounds__(256) void mod_kernel(
    const float* __restrict__ hactiv, const float* __restrict__ h2mod_w,
    const float* __restrict__ h2mod_b, const float* __restrict__ mfw,
    const float* __restrict__ mfb, const float* __restrict__ ws_itc,
    float* __restrict__ ws_g)
{
    __shared__ float red[256];
    __shared__ float mval;
    int b   = blockIdx.x;
    int tid = threadIdx.x;
    size_t base = (size_t)b * HSZ;
    int k0 = tid << 1;
    red[tid] = hactiv[base + k0] * h2mod_w[k0]
             + hactiv[base + k0 + 1] * h2mod_w[k0 + 1];
    __syncthreads();
    for (int off = 128; off > 0; off >>= 1) {
        if (tid < off) red[tid] += red[tid + off];
        __syncthreads();
    }
    if (tid == 0) mval = tanhf(red[0] + h2mod_b[0]);
    __syncthreads();
    float m = mval;
#pragma unroll
    for (int j = 0; j < 2; ++j) {
        int k = k0 + j;
        ws_g[base + k] = (m * mfw[k] + mfb[k]) * ws_itc[base + k];
    }
}

// ---------------------------------------------------------------------------
// Kernel 4: hebb_new = clip(hebb + h0[b][h]*g[b][k], +-2)  (536 MB traffic)
// 2 rows per 256-thread block; non-temporal B128 both directions.
// ---------------------------------------------------------------------------
__global__ __launch_bounds__(256) void hebb_update_kernel(
    const float* __restrict__ h0, const float* __restrict__ hebb,
    const float* __restrict__ ws_g, float* __restrict__ hebb_out)
{
    int row = blockIdx.x * 2 + (threadIdx.x >> 7);   // row = b*HSZ + h
    int tid = threadIdx.x & 127;                     // 128 float4 per row
    int b   = row >> 9;
    float hv = h0[row];
    const v4f* hp = (const v4f*)(hebb + (size_t)row * HSZ);
    const v4f* gp = (const v4f*)(ws_g + (size_t)b * HSZ);
    v4f*       op = (v4f*)(hebb_out + (size_t)row * HSZ);

    v4f hb4 = __builtin_nontemporal_load(hp + tid);
    v4f g4  = gp[tid];
    v4f r;
#pragma unroll
    for (int c = 0; c < 4; ++c)
        r[c] = fminf(fmaxf(hb4[c] + hv * g4[c], -CLIPV), CLIPV);
    __builtin_nontemporal_store(r, op + tid);
}

// ---------------------------------------------------------------------------
extern "C" void kernel_launch(void* const* d_in, const int* in_sizes, int n_in,
                              void* d_out, int out_size, void* d_ws, size_t ws_size,
                              hipStream_t stream) {
    const float* x     = (const float*)d_in[0];
    const float* h0    = (const float*)d_in[1];
    const float* c0    = (const float*)d_in[2];
    const float* hebb  = (const float*)d_in[3];
    const float* w     = (const float*)d_in[4];
    const float* alpha = (const float*)d_in[5];
    const float* h2f_w = (const float*)d_in[6];
    const float* h2f_b = (const float*)d_in[7];
    const float* h2i_w = (const float*)d_in[8];
    const float* h2i_b = (const float*)d_in[9];
    const float* h2o_w = (const float*)d_in[10];
    const float* h2o_b = (const float*)d_in[11];
    const float* x2f_w = (const float*)d_in[12];
    const float* x2f_b = (const float*)d_in[13];
    const float* x2i_w = (const float*)d_in[14];
    const float* x2i_b = (const float*)d_in[15];
    const float* x2o_w = (const float*)d_in[16];
    const float* x2o_b = (const float*)d_in[17];
    const float* x2c_w = (const float*)d_in[18];
    const float* x2c_b = (const float*)d_in[19];
    const float* h2mod_w = (const float*)d_in[20];
    const float* h2mod_b = (const float*)d_in[21];
    const float* mfw     = (const float*)d_in[22];
    const float* mfb     = (const float*)d_in[23];

    float* out      = (float*)d_out;
    float* hactiv   = out;                  // (256,512)
    float* cell     = out + BH;             // (256,512)
    float* hebb_out = out + 2 * (size_t)BH; // (256,512,512)

    float* ws      = (float*)d_ws;          // jobs 0..4 at offsets 0..4*BH
    float* ws_itc  = ws + 5 * (size_t)BH;
    float* ws_g    = ws + 6 * (size_t)BH;
    float* ws_part = ws + 7 * (size_t)BH;   // NCHUNK*BH partial sums (5.8 MB total)

    // 1) GEMMs: 5 jobs * 512 tiles = 2560 waves = 320 blocks of 8 waves
    gate_gemm_kernel<<<320, 256, 0, stream>>>(
        x, h0, w,
        h2f_w, h2f_b, h2i_w, h2i_b, h2o_w, h2o_b,
        x2f_w, x2f_b, x2i_w, x2i_b, x2o_w, x2o_b, x2c_w, x2c_b, ws);

    // 2a) hebb reduction partials: 1024 blocks streaming 268 MB (B128, NT)
    plastic_partial_kernel<<<dim3(BATCH, NCHUNK), 128, 0, stream>>>(
        h0, hebb, ws_part);

    // 2b) combine + LSTM cell fusion
    plastic_combine_kernel<<<BATCH, 128, 0, stream>>>(
        c0, alpha, ws, ws_part, ws_itc, hactiv, cell);

    // 3) neuromodulator + per-column update gain
    mod_kernel<<<BATCH, 256, 0, stream>>>(
        hactiv, h2mod_w, h2mod_b, mfw, mfb, ws_itc, ws_g);

    // 4) streaming hebb update (B128, NT both directions)
    hebb_update_kernel<<<BATCH * HSZ / 2, 256, 0, stream>>>(
        h0, hebb, ws_g, hebb_out);
}